// ChamferDistanceLoss_50070728737514
// MI455X (gfx1250) — compile-verified
//
#include <hip/hip_runtime.h>

typedef __attribute__((ext_vector_type(2))) float v2f;
typedef __attribute__((ext_vector_type(8))) float v8f;

#define NPTS  8192
#define TILE  16
#define WPB   4      // waves per block
#define BATCH 4

// ---------------------------------------------------------------------------
// Prep: pack each point p=(x,y,z) as (-2x, -2y, -2z, |p|^2) so the hot loop
// needs only a b128 load + 2 cndmask selects to build the WMMA B operand.
// ---------------------------------------------------------------------------
__global__ void pack_points_kernel(const float* __restrict__ tmpl,
                                   const float* __restrict__ srcp,
                                   float4* __restrict__ t4,
                                   float4* __restrict__ s4)
{
    const int i = blockIdx.x * blockDim.x + threadIdx.x;   // 0 .. BATCH*NPTS-1
    if (i >= BATCH * NPTS) return;
    {
        float x = tmpl[3 * i + 0], y = tmpl[3 * i + 1], z = tmpl[3 * i + 2];
        t4[i] = make_float4(-2.0f * x, -2.0f * y, -2.0f * z, x * x + y * y + z * z);
    }
    {
        float x = srcp[3 * i + 0], y = srcp[3 * i + 1], z = srcp[3 * i + 2];
        s4[i] = make_float4(-2.0f * x, -2.0f * y, -2.0f * z, x * x + y * y + z * z);
    }
}

// ---------------------------------------------------------------------------
// Main (packed): one wave owns a 16-row tile, sweeps all 512 column tiles
// with V_WMMA_F32_16X16X4_F32 (K=4 = x,y,z + augmented slot), running min.
// D = A x B + C with A[n]=(x,y,z,1), B[:,m]=(-2x,-2y,-2z,|p|^2), C[n][m]=|row n|^2
// gives the full squared-distance tile in a single WMMA.
// ---------------------------------------------------------------------------
__global__ void __launch_bounds__(WPB * 32)
chamfer_wmma_packed(const float4* __restrict__ t4,
                    const float4* __restrict__ s4,
                    float* __restrict__ out)
{
    const int lane = threadIdx.x & 31;
    const int wave = threadIdx.x >> 5;
    const int b    = blockIdx.y;

    const float4* rp = (blockIdx.z ? s4 : t4) + (size_t)b * NPTS;
    const float4* cp = (blockIdx.z ? t4 : s4) + (size_t)b * NPTS;

    const int rowBase = (blockIdx.x * WPB + wave) * TILE;
    const int half = lane >> 4;   // 0: lanes 0-15 (K=0,1), 1: lanes 16-31 (K=2,3)
    const int l16  = lane & 15;

    // A operand: recover raw coords from packed (-2x,-2y,-2z,sq) once.
    float4 p = rp[rowBase + l16];
    float ax = -0.5f * p.x, ay = -0.5f * p.y, az = -0.5f * p.z;
    v2f A;
    A.x = half ? az   : ax;       // K = 2*half
    A.y = half ? 1.0f : ay;       // K = 2*half + 1 (passes B's sq row through)

    // C: C[r] = |row (r + 8*half)|^2  (row i's sq lives in lane i)
    v8f C;
#pragma unroll
    for (int r = 0; r < 8; ++r)
        C[r] = __shfl(p.w, r + 8 * half, 32);

    v8f acc;
#pragma unroll
    for (int r = 0; r < 8; ++r) acc[r] = 3.402823466e38f;

    for (int t = 0; t < NPTS / TILE; ++t) {
        float4 q = cp[t * TILE + l16];
        v2f Bv;
        Bv.x = half ? q.z : q.x;  // (-2z | -2x)
        Bv.y = half ? q.w : q.y;  // ( sq | -2y)

        v8f D = __builtin_amdgcn_wmma_f32_16x16x4_f32(
            false, A, false, Bv, (short)0, C, false, false);

#pragma unroll
        for (int r = 0; r < 8; ++r)
            acc[r] = fminf(acc[r], D[r]);
    }

    // min across each 16-lane half (columns sharing a row)
#pragma unroll
    for (int off = 1; off < 16; off <<= 1) {
#pragma unroll
        for (int r = 0; r < 8; ++r)
            acc[r] = fminf(acc[r], __shfl_xor(acc[r], off, 32));
    }

    float s = 0.0f;
#pragma unroll
    for (int r = 0; r < 8; ++r) {
        float d = acc[r] > 0.0f ? acc[r] : 0.0f;
        s += sqrtf(d);
    }
    s += __shfl_xor(s, 16, 32);   // rows 0..7 (lanes<16) + rows 8..15 (lanes>=16)

    if (lane == 0)
        atomicAdd(out, s * (1.0f / 65536.0f));  // / (8192 * 2 directions * 4 batches)
}

// ---------------------------------------------------------------------------
// Fallback (no workspace): identical math, loads raw xyz and builds operands
// in the loop.
// ---------------------------------------------------------------------------
__global__ void __launch_bounds__(WPB * 32)
chamfer_wmma_direct(const float* __restrict__ tmpl,
                    const float* __restrict__ srcp,
                    float* __restrict__ out)
{
    const int lane = threadIdx.x & 31;
    const int wave = threadIdx.x >> 5;
    const int b    = blockIdx.y;

    const float* rp = (blockIdx.z ? srcp : tmpl) + (size_t)b * NPTS * 3;
    const float* cp = (blockIdx.z ? tmpl : srcp) + (size_t)b * NPTS * 3;

    const int rowBase = (blockIdx.x * WPB + wave) * TILE;
    const int half = lane >> 4;
    const int l16  = lane & 15;

    const int rrow = rowBase + l16;
    float ax = rp[3 * rrow + 0], ay = rp[3 * rrow + 1], az = rp[3 * rrow + 2];
    float asq = ax * ax + ay * ay + az * az;

    v2f A;
    A.x = half ? az   : ax;
    A.y = half ? 1.0f : ay;

    v8f C;
#pragma unroll
    for (int r = 0; r < 8; ++r)
        C[r] = __shfl(asq, r + 8 * half, 32);

    v8f acc;
#pragma unroll
    for (int r = 0; r < 8; ++r) acc[r] = 3.402823466e38f;

    for (int t = 0; t < NPTS / TILE; ++t) {
        const int m = t * TILE + l16;
        float bx = cp[3 * m + 0], by = cp[3 * m + 1], bz = cp[3 * m + 2];
        float bsq = bx * bx + by * by + bz * bz;
        v2f Bv;
        Bv.x = half ? (-2.0f * bz) : (-2.0f * bx);
        Bv.y = half ? bsq          : (-2.0f * by);

        v8f D = __builtin_amdgcn_wmma_f32_16x16x4_f32(
            false, A, false, Bv, (short)0, C, false, false);

#pragma unroll
        for (int r = 0; r < 8; ++r)
            acc[r] = fminf(acc[r], D[r]);
    }

#pragma unroll
    for (int off = 1; off < 16; off <<= 1) {
#pragma unroll
        for (int r = 0; r < 8; ++r)
            acc[r] = fminf(acc[r], __shfl_xor(acc[r], off, 32));
    }

    float s = 0.0f;
#pragma unroll
    for (int r = 0; r < 8; ++r) {
        float d = acc[r] > 0.0f ? acc[r] : 0.0f;
        s += sqrtf(d);
    }
    s += __shfl_xor(s, 16, 32);

    if (lane == 0)
        atomicAdd(out, s * (1.0f / 65536.0f));
}

extern "C" void kernel_launch(void* const* d_in, const int* in_sizes, int n_in,
                              void* d_out, int out_size, void* d_ws, size_t ws_size,
                              hipStream_t stream) {
    const float* tmpl = (const float*)d_in[0];   // [4, 8192, 3] fp32
    const float* srcp = (const float*)d_in[1];   // [4, 8192, 3] fp32
    float* out = (float*)d_out;                  // scalar fp32

    hipMemsetAsync(out, 0, sizeof(float), stream);

    dim3 block(WPB * 32);
    dim3 grid((NPTS / TILE) / WPB, BATCH, 2 /*directions*/);

    const size_t needed = (size_t)2 * BATCH * NPTS * sizeof(float4);  // 1 MB
    if (ws_size >= needed) {
        float4* t4 = (float4*)d_ws;
        float4* s4 = t4 + (size_t)BATCH * NPTS;
        const int npts = BATCH * NPTS;
        pack_points_kernel<<<(npts + 255) / 256, 256, 0, stream>>>(tmpl, srcp, t4, s4);
        chamfer_wmma_packed<<<grid, block, 0, stream>>>(t4, s4, out);
    } else {
        chamfer_wmma_direct<<<grid, block, 0, stream>>>(tmpl, srcp, out);
    }
}